// Capsule_18193481465979
// MI455X (gfx1250) — compile-verified
//
#include <hip/hip_runtime.h>
#include <math.h>

// Problem constants (from reference)
#define N_CAPS   64
#define CAPS_DIM 16
#define OUT_DIM  128
#define PDIM     1152
#define BATCH    256
#define BO       (BATCH * OUT_DIM)   // 32768
#define PCH      8                   // p-chunks for partial reduction
#define PPER     (PDIM / PCH)        // 144

// Workspace layout (in floats)
#define WS_PART  0                   // PCH * BO = 262144 floats
#define WS_XSUM  (PCH * BO)          // BO floats
#define WS_WSUM  (WS_XSUM + BO)      // N_CAPS*OUT_DIM = 8192 floats
// total = 303104 floats = 1.16 MB

typedef float v2f __attribute__((ext_vector_type(2)));
typedef float v8f __attribute__((ext_vector_type(8)));

// ---------------------------------------------------------------------------
// Kernel 1: Xsum partials.  xp factorizes: xp[b,n,o] = Wsum[n,o]*Xsum[b,o].
// Reading x (151 MB) once is the roofline floor (~6.5us @ 23.3 TB/s).
// Thread t of block (bx,pc) accumulates x[p, e/128, e%128] over 144 p values.
// Consecutive threads hit consecutive addresses -> fully coalesced.
// ---------------------------------------------------------------------------
__global__ __launch_bounds__(256) void k_reduce_x(const float* __restrict__ x,
                                                  float* __restrict__ ws) {
  const int e  = blockIdx.x * 256 + threadIdx.x;   // 0..BO-1
  const int pc = blockIdx.y;                       // 0..PCH-1
  const float* px = x + (size_t)pc * PPER * BO + e;
  float s = 0.0f;
#pragma unroll 8
  for (int p = 0; p < PPER; ++p) s += px[(size_t)p * BO];
  ws[WS_PART + pc * BO + e] = s;
}

// ---------------------------------------------------------------------------
// Kernel 2: fold partials -> Xsum; reduce caps_weights over c -> Wsum.
// ---------------------------------------------------------------------------
__global__ __launch_bounds__(256) void k_finalize(const float* __restrict__ w,
                                                  float* __restrict__ ws) {
  const int t = threadIdx.x;
  if (blockIdx.x < BO / 256) {
    const int e = blockIdx.x * 256 + t;
    float s = 0.0f;
#pragma unroll
    for (int pc = 0; pc < PCH; ++pc) s += ws[WS_PART + pc * BO + e];
    ws[WS_XSUM + e] = s;
  } else {
    const int idx = (blockIdx.x - BO / 256) * 256 + t;  // 0..8191
    const int n = idx >> 7, o = idx & 127;
    float s = 0.0f;
#pragma unroll
    for (int c = 0; c < CAPS_DIM; ++c) s += w[(n * CAPS_DIM + c) * OUT_DIM + o];
    ws[WS_WSUM + idx] = s;
  }
}

// ---------------------------------------------------------------------------
// fp32 WMMA wrapper: D(16x16) = A(16x4) x B(4x16) + C
// ---------------------------------------------------------------------------
__device__ inline v8f wmma_f32(v2f a, v2f b, v8f c) {
  return __builtin_amdgcn_wmma_f32_16x16x4_f32(false, a, false, b,
                                               (short)0, c, false, false);
}

__device__ inline float waveSum(float v) {
#pragma unroll
  for (int m = 16; m >= 1; m >>= 1) v += __shfl_xor(v, m, 32);
  return v;
}
__device__ inline float waveMax(float v) {
#pragma unroll
  for (int m = 16; m >= 1; m >>= 1) v = fmaxf(v, __shfl_xor(v, m, 32));
  return v;
}

// ---------------------------------------------------------------------------
// Kernel 3: dynamic routing, one wave32 per batch element.
//
// WMMA operand layouts (CDNA5 ISA 7.12.2, wave32):
//   A(16x4)  f32: lane l -> M = l%16 ; VGPR v, half(l>=16) -> K = v + 2*(l/16)
//   B(4x16)  f32: lane l -> N = l%16 ; VGPR v, half(l>=16) -> K = v + 2*(l/16)
//   C/D(16x16)  : lane l -> N = l%16 ; VGPR j -> M = j + 8*(l/16)
//
// routed[o] = sum_n cf[n]*xp[n][o]:   A = cf replicated over M (K=n chunk),
//                                     B[k][n] = xp[n-chunk k][o-tile col n]
//                                     -> every lane/VGPR of D = routed[o-tile, N]
// logits[n] = sum_o r[o]*xp[n][o]:    A[m][k] = xp[n-tile m][o-chunk k],
//                                     B = r replicated over N
//                                     -> D[j] = logits[n-tile, j+8*hi], any N
// ---------------------------------------------------------------------------
__global__ __launch_bounds__(32) void k_routing(const float* __restrict__ ws,
                                                float* __restrict__ out) {
  __shared__ float xpL[N_CAPS * OUT_DIM];  // 32 KB
  __shared__ float rS[OUT_DIM];
  __shared__ float lgS[N_CAPS];
  __shared__ float cfS[N_CAPS];

  const int b    = blockIdx.x;
  const int lane = threadIdx.x;
  const int lm   = lane & 15;
  const int hi   = lane >> 4;  // 0 or 1

  const float* Xs = ws + WS_XSUM + b * OUT_DIM;
  const float* Wm = ws + WS_WSUM;

  // xp[n][o] = Wsum[n][o] * Xsum[b][o]
  float xv[4];
#pragma unroll
  for (int j = 0; j < 4; ++j) xv[j] = Xs[lane + 32 * j];
#pragma unroll 4
  for (int k = 0; k < 256; ++k) {
    const int i = lane + 32 * k;           // i%128 == lane + 32*(k%4)
    xpL[i] = Wm[i] * xv[k & 3];
  }
  cfS[lane]      = 1.0f / 64.0f;           // softmax(zeros)
  cfS[lane + 32] = 1.0f / 64.0f;
  lgS[lane]      = 0.0f;
  lgS[lane + 32] = 0.0f;
  __syncthreads();

  for (int it = 0; it < 3; ++it) {
    // ---- routed[o] = sum_n cf[n] * xp[n][o]  (8 o-tiles x 16 K-chunks) ----
    v2f av[16];
#pragma unroll
    for (int ch = 0; ch < 16; ++ch) {
      av[ch].x = cfS[4 * ch + 2 * hi];
      av[ch].y = cfS[4 * ch + 2 * hi + 1];
    }
#pragma unroll
    for (int ot = 0; ot < 8; ++ot) {
      v8f acc = {0.f, 0.f, 0.f, 0.f, 0.f, 0.f, 0.f, 0.f};
#pragma unroll
      for (int ch = 0; ch < 16; ++ch) {
        const int row = 4 * ch + 2 * hi;
        v2f bv;
        bv.x = xpL[row * OUT_DIM + ot * 16 + lm];
        bv.y = xpL[(row + 1) * OUT_DIM + ot * 16 + lm];
        acc = wmma_f32(av[ch], bv, acc);
      }
      if (hi == 0) rS[ot * 16 + lm] = acc[0];  // all rows equal; M=0 lane copy
    }
    __syncthreads();

    // ---- squash scale over O ----
    float rv[4], p0 = 0.0f;
#pragma unroll
    for (int j = 0; j < 4; ++j) {
      rv[j] = rS[lane + 32 * j];
      p0 += rv[j] * rv[j];
    }
    const float nrm2  = waveSum(p0);
    const float nrm   = sqrtf(nrm2);
    const float scale = nrm / (1.0f + nrm2);

    if (it == 2) {
      // final output: squash(coeffs @ xp)
#pragma unroll
      for (int j = 0; j < 4; ++j)
        out[b * OUT_DIM + lane + 32 * j] = rv[j] * scale;
    } else {
#pragma unroll
      for (int j = 0; j < 4; ++j) rS[lane + 32 * j] = rv[j] * scale;
      __syncthreads();

      // ---- logits[n] += sum_o r[o] * xp[n][o]  (4 n-tiles x 32 K-chunks) ----
#pragma unroll
      for (int nt = 0; nt < 4; ++nt) {
        v8f acc = {0.f, 0.f, 0.f, 0.f, 0.f, 0.f, 0.f, 0.f};
#pragma unroll
        for (int ch = 0; ch < 32; ++ch) {
          const int k0 = 4 * ch + 2 * hi;
          v2f a2, b2;
          a2.x = xpL[(nt * 16 + lm) * OUT_DIM + k0];
          a2.y = xpL[(nt * 16 + lm) * OUT_DIM + k0 + 1];
          b2.x = rS[k0];
          b2.y = rS[k0 + 1];
          acc = wmma_f32(a2, b2, acc);
        }
        if (lm == 0) {  // lanes 0 and 16 cover M = j and M = j+8
#pragma unroll
          for (int j = 0; j < 8; ++j) lgS[nt * 16 + 8 * hi + j] += acc[j];
        }
      }
      __syncthreads();

      // ---- coeffs = softmax(logits) over N=64 ----
      const float l0 = lgS[lane], l1 = lgS[lane + 32];
      const float mx = waveMax(fmaxf(l0, l1));
      const float e0 = expf(l0 - mx), e1 = expf(l1 - mx);
      const float sm = waveSum(e0 + e1);
      cfS[lane]      = e0 / sm;
      cfS[lane + 32] = e1 / sm;
      __syncthreads();
    }
  }
}

// ---------------------------------------------------------------------------
extern "C" void kernel_launch(void* const* d_in, const int* in_sizes, int n_in,
                              void* d_out, int out_size, void* d_ws, size_t ws_size,
                              hipStream_t stream) {
  (void)in_sizes; (void)n_in; (void)out_size; (void)ws_size;
  const float* x = (const float*)d_in[0];        // (1152, 256, 128) f32
  const float* w = (const float*)d_in[1];        // (64, 16, 128) f32
  float* out = (float*)d_out;                    // (256, 1, 128) f32
  float* ws  = (float*)d_ws;

  dim3 g1(BO / 256, PCH);
  k_reduce_x<<<g1, 256, 0, stream>>>(x, ws);
  k_finalize<<<BO / 256 + (N_CAPS * OUT_DIM) / 256, 256, 0, stream>>>(w, ws);
  k_routing<<<BATCH, 32, 0, stream>>>(ws, out);
}